// Transform_51067161149510
// MI455X (gfx1250) — compile-verified
//
#include <hip/hip_runtime.h>

// Homography warp + bilinear sampling, B=32, C=3, H=W=512, fp32.
// Memory-bound gather kernel: ~100MB read + ~100MB written -> ~8.6us floor at
// 23.3 TB/s on MI455X. WMMA does not apply (no dense GEMM structure); optimize
// the data-movement path instead:
//   - one full image row per 256-thread block, 2 pixels/thread (x, x+256):
//     amortizes the per-wave scalar preamble (workgroup-id decode, H fetch,
//     scalar-folded y terms) over 64 pixels/wave, lanes stay x-contiguous so
//     every gather is a row-contiguous 128B wave footprint
//   - y is wave-uniform -> compiler scalarizes yt math (s_fmac_f32 path)
//   - NT stores: output streams past L2 so the 100MB image stays L2-resident
//   - v_rcp_f32 + Newton instead of the 9-instruction IEEE div expansion

namespace {

constexpr int kB  = 32;
constexpr int kC  = 3;
constexpr int kH  = 512;
constexpr int kW  = 512;
constexpr int kHW = kH * kW;

__global__ __launch_bounds__(256) void homography_bilinear_warp(
    const float* __restrict__ img,   // [B, C, H, W]
    const float* __restrict__ Hmat,  // [B, 9]
    float* __restrict__ out)         // [B, C, H, W]
{
    const int b   = blockIdx.y;      // wave-uniform -> s_load for H
    const int y   = blockIdx.x;      // one full row per block (wave-uniform)
    const int tid = threadIdx.x;

    // 3x3 homography, scalar (uniform) loads
    const float* Hb = Hmat + b * 9;
    const float h0 = Hb[0], h1 = Hb[1], h2 = Hb[2];
    const float h3 = Hb[3], h4 = Hb[4], h5 = Hb[5];
    const float h6 = Hb[6], h7 = Hb[7], h8 = Hb[8];

    // normalized coords in [-1, 1] (linspace semantics: -1 + i * 2/(N-1))
    const float sx = 2.0f / (float)(kW - 1);
    const float sy = 2.0f / (float)(kH - 1);
    const float yt = fmaf((float)y, sy, -1.0f);          // scalarized

    // y-dependent parts of T = H * [xt, yt, 1] (wave-uniform -> s_fmac)
    const float c0 = fmaf(h1, yt, h2);
    const float c1 = fmaf(h4, yt, h5);
    const float c2 = fmaf(h7, yt, h8);

    const float* imgb  = img + b * (kC * kHW);
    const int    rbase = (b * kC) * kHW + y * kW;

#pragma unroll
    for (int k = 0; k < 2; ++k) {
        const int   x  = tid + k * 256;
        const float xt = fmaf((float)x, sx, -1.0f);

        const float T0 = fmaf(h0, xt, c0);
        const float T1 = fmaf(h3, xt, c1);
        const float T2 = fmaf(h6, xt, c2);

        // perspective divide: v_rcp_f32 + one Newton-Raphson step (~1 ulp)
        float r = __builtin_amdgcn_rcpf(T2);
        r = fmaf(fmaf(-T2, r, 1.0f), r, r);

        // normalized -> pixel coords: (v + 1) * (N/2)
        const float xs = fmaf(T0 * r, (float)(kW / 2), (float)(kW / 2));
        const float ys = fmaf(T1 * r, (float)(kH / 2), (float)(kH / 2));

        const float fx = floorf(xs);
        const float fy = floorf(ys);
        const float x0 = fminf(fmaxf(fx,        0.0f), (float)(kW - 1));
        const float x1 = fminf(fmaxf(fx + 1.0f, 0.0f), (float)(kW - 1));
        const float y0 = fminf(fmaxf(fy,        0.0f), (float)(kH - 1));
        const float y1 = fminf(fmaxf(fy + 1.0f, 0.0f), (float)(kH - 1));

        // Weights use the CLAMPED corners vs the UNCLAMPED sample position,
        // exactly like the reference (this zeroes fully-clamped regions).
        const float wa = (x1 - xs) * (y1 - ys);
        const float wb = (x1 - xs) * (ys - y0);
        const float wc = (xs - x0) * (y1 - ys);
        const float wd = (xs - x0) * (ys - y0);

        const int ix0 = (int)x0, ix1 = (int)x1;
        const int iy0 = (int)y0, iy1 = (int)y1;
        const int ia = iy0 * kW + ix0;
        const int ib = iy1 * kW + ix0;
        const int ic = iy0 * kW + ix1;
        const int id = iy1 * kW + ix1;

#pragma unroll
        for (int c = 0; c < kC; ++c) {
            const float* p  = imgb + c * kHW;
            const float Ia = p[ia];   // temporal gathers: stay in L2
            const float Ib = p[ib];
            const float Ic = p[ic];
            const float Id = p[id];
            const float v  = fmaf(wa, Ia, fmaf(wb, Ib, fmaf(wc, Ic, wd * Id)));
            // write-once output: non-temporal, streams past L2
            __builtin_nontemporal_store(v, out + rbase + c * kHW + x);
        }
    }
}

} // namespace

extern "C" void kernel_launch(void* const* d_in, const int* in_sizes, int n_in,
                              void* d_out, int out_size, void* d_ws, size_t ws_size,
                              hipStream_t stream) {
    (void)in_sizes; (void)n_in; (void)d_ws; (void)ws_size; (void)out_size;
    const float* img  = (const float*)d_in[0];   // image2_tensor [32,3,512,512] f32
    const float* Hmat = (const float*)d_in[1];   // H_tf [32,9] f32
    float* out = (float*)d_out;                  // [32,3,512,512] f32

    dim3 grid(kH, kB);          // (512, 32): one image row per block
    dim3 block(256);            // 8 wave32 waves, 2 px/thread
    homography_bilinear_warp<<<grid, block, 0, stream>>>(img, Hmat, out);
}